// TransformerModel_12979391169241
// MI455X (gfx1250) — compile-verified
//
#include <hip/hip_runtime.h>
#include <hip/hip_bf16.h>
#include <math.h>

#define DD 128           // model dim
#define LL 128           // sequence length
#define BATCH 16
#define NLAYERS 2
#define NUM_POI 5000
#define NUM_CAT 300
#define ROWS (BATCH * LL)        // 2048
#define NEGV (-4294967295.0f)    // -2^32 + 1
#define EPSV 1e-8f
#define SCALEV 0.08838834764831843f  // 1/sqrt(128)

typedef float v2f __attribute__((ext_vector_type(2)));
typedef float v8f __attribute__((ext_vector_type(8)));

// V_WMMA_F32_16X16X4_F32:  D(16x16 f32) = A(16x4 f32) * B(4x16 f32) + C
__device__ __forceinline__ v8f wmma_f32(v2f a, v2f b, v8f c) {
    return __builtin_amdgcn_wmma_f32_16x16x4_f32(
        /*neg_a=*/false, a, /*neg_b=*/false, b,
        /*c_mod=*/(short)0, c, /*reuse_a=*/false, /*reuse_b=*/false);
}

__device__ __forceinline__ float wave_sum(float s) {
    s += __shfl_xor(s, 16, 32);
    s += __shfl_xor(s, 8, 32);
    s += __shfl_xor(s, 4, 32);
    s += __shfl_xor(s, 2, 32);
    s += __shfl_xor(s, 1, 32);
    return s;
}
__device__ __forceinline__ float wave_max(float s) {
    s = fmaxf(s, __shfl_xor(s, 16, 32));
    s = fmaxf(s, __shfl_xor(s, 8, 32));
    s = fmaxf(s, __shfl_xor(s, 4, 32));
    s = fmaxf(s, __shfl_xor(s, 2, 32));
    s = fmaxf(s, __shfl_xor(s, 1, 32));
    return s;
}

// ---------------- LayerNorm: one wave (32 lanes) per row of 128 --------------
template <bool HAS_ADD>
__global__ void ln_kernel(const float* __restrict__ x,
                          const float* __restrict__ g,
                          const float* __restrict__ b,
                          const float* __restrict__ add,
                          float* __restrict__ out) {
    int row  = blockIdx.x;
    int lane = threadIdx.x;
    const float4* xr = (const float4*)(x + (size_t)row * DD);
    float4 v = xr[lane];
    float mean = wave_sum(v.x + v.y + v.z + v.w) * (1.0f / DD);
    float4 d;
    d.x = v.x - mean; d.y = v.y - mean; d.z = v.z - mean; d.w = v.w - mean;
    float var = wave_sum(d.x*d.x + d.y*d.y + d.z*d.z + d.w*d.w) * (1.0f / DD);
    float inv = rsqrtf(var + EPSV);
    float4 gg = ((const float4*)g)[lane];
    float4 bb = ((const float4*)b)[lane];
    float4 o;
    o.x = d.x * inv * gg.x + bb.x;
    o.y = d.y * inv * gg.y + bb.y;
    o.z = d.z * inv * gg.z + bb.z;
    o.w = d.w * inv * gg.w + bb.w;
    if (HAS_ADD) {
        float4 a = ((const float4*)(add + (size_t)row * DD))[lane];
        o.x += a.x; o.y += a.y; o.z += a.z; o.w += a.w;
    }
    ((float4*)(out + (size_t)row * DD))[lane] = o;
}

// ---------------- Softmax over last dim (128), one wave per row --------------
__global__ void softmax_kernel(float* __restrict__ scores) {
    int row  = blockIdx.x;
    int lane = threadIdx.x;
    float4* sr = (float4*)(scores + (size_t)row * LL);
    float4 v = sr[lane];
    float mx = wave_max(fmaxf(fmaxf(v.x, v.y), fmaxf(v.z, v.w)));
    v.x = __expf(v.x - mx); v.y = __expf(v.y - mx);
    v.z = __expf(v.z - mx); v.w = __expf(v.w - mx);
    float s = wave_sum(v.x + v.y + v.z + v.w);
    float inv = 1.0f / s;
    v.x *= inv; v.y *= inv; v.z *= inv; v.w *= inv;
    sr[lane] = v;
}

// ---------------- Generic WMMA GEMM:  out(2048 x N) = A(2048 x 128) @ B(128 x N)
// Compile-time config -> branch-free epilogue, immediate-offset loads.
// Double-buffered fragment staging: loads for chunk c+1 issue before the WMMA
// chain of chunk c, so s_wait_loadcnt counts down instead of stalling at 0.
template <int N, bool BIAS, bool RELU, bool RESID, bool KEEP, bool BATCHED, bool STREAM>
__global__ void gemm_kernel(const float* __restrict__ A,
                            const float* __restrict__ Bw,
                            const float* __restrict__ bias,
                            const float* __restrict__ resid,
                            const int*   __restrict__ rowmask,
                            float* __restrict__ out) {
    int lane = threadIdx.x & 31;
    int wave = threadIdx.x >> 5;
    int m0 = blockIdx.x * 16;
    int n0 = blockIdx.y * 128 + wave * 16;
    if (n0 >= N) return;                 // uniform per-wave exit (tail blocks)
    int hi = lane >> 4;                  // half-wave select
    int lm = lane & 15;
    int nn = n0 + lm;
    int nc = nn < N ? nn : N - 1;        // clamp for tail-tile loads
    const float* Bp = Bw + (BATCHED ? (size_t)(blockIdx.x >> 3) * DD * N : (size_t)0);
    const float* bcol = Bp + nc + 2 * hi * N;  // one base; k*N is an immediate
    const float* arow = A + (size_t)(m0 + lm) * DD + 2 * hi;

    v8f acc = {};
    if (BIAS) {
        float binit = bias[nc];
        #pragma unroll
        for (int r = 0; r < 8; ++r) acc[r] = binit;
    }

    constexpr int CH = 8;                      // k-steps per chunk
    constexpr int NCHUNK = DD / (4 * CH);      // 4 chunks of 8 WMMAs
    v2f abuf[2][CH], bbuf[2][CH];

    #pragma unroll
    for (int j = 0; j < CH; ++j) {             // prologue: stage chunk 0
        const int k = 4 * j;
        abuf[0][j] = *(const v2f*)(arow + k);
        bbuf[0][j].x = bcol[k * N];
        bbuf[0][j].y = bcol[(k + 1) * N];
    }
    #pragma unroll
    for (int c = 0; c < NCHUNK; ++c) {
        const int cur = c & 1, nxt = cur ^ 1;
        if (c + 1 < NCHUNK) {                  // stage chunk c+1 while computing c
            #pragma unroll
            for (int j = 0; j < CH; ++j) {
                const int k = 4 * (CH * (c + 1) + j);
                abuf[nxt][j] = *(const v2f*)(arow + k);
                bbuf[nxt][j].x = bcol[k * N];
                bbuf[nxt][j].y = bcol[(k + 1) * N];
            }
        }
        #pragma unroll
        for (int j = 0; j < CH; ++j)
            acc = wmma_f32(abuf[cur][j], bbuf[cur][j], acc);
    }

    // Epilogue: branch-free (all template-selected)
    const int mbase = m0 + 8 * hi;
    #pragma unroll
    for (int r = 0; r < 8; ++r) {
        int m = mbase + r;                     // C/D: M = r + 8*hi
        float val = acc[r];
        if (RELU) val = fmaxf(val, 0.0f);
        if (nn < N) {
            if (RESID) val += resid[(size_t)m * N + nn];
            if (KEEP)  val = rowmask[m] ? 0.0f : val;   // keep = !time_mask
            if (STREAM) __builtin_nontemporal_store(val, &out[(size_t)m * N + nn]);
            else        out[(size_t)m * N + nn] = val;
        }
    }
}

// ---------------- Scores: qk^T (WMMA) + time-matrix term + scale + mask ------
// grid (B, L/16); block 256 (8 waves), wave w handles key-tile n0 = 16*w.
__global__ void scores_kernel(const float* __restrict__ q,
                              const float* __restrict__ kmat,
                              const float* __restrict__ tm,
                              const int*   __restrict__ tmask,
                              float* __restrict__ scores) {
    int bIdx = blockIdx.x;
    int m0 = blockIdx.y * 16;
    int lane = threadIdx.x & 31;
    int wave = threadIdx.x >> 5;
    int n0 = wave * 16;
    int hi = lane >> 4;
    int lm = lane & 15;

    const float* qb = q    + (size_t)bIdx * LL * DD;
    const float* kb = kmat + (size_t)bIdx * LL * DD;
    const float* arow = qb + (size_t)(m0 + lm) * DD + 2 * hi;
    const float* brow = kb + (size_t)(n0 + lm) * DD + 2 * hi; // B = k^T -> contiguous

    v8f acc = {};
    constexpr int CH = 8;
    constexpr int NCHUNK = DD / (4 * CH);
    v2f abuf[2][CH], bbuf[2][CH];
    #pragma unroll
    for (int j = 0; j < CH; ++j) {
        const int k = 4 * j;
        abuf[0][j] = *(const v2f*)(arow + k);
        bbuf[0][j] = *(const v2f*)(brow + k);
    }
    #pragma unroll
    for (int c = 0; c < NCHUNK; ++c) {
        const int cur = c & 1, nxt = cur ^ 1;
        if (c + 1 < NCHUNK) {
            #pragma unroll
            for (int j = 0; j < CH; ++j) {
                const int k = 4 * (CH * (c + 1) + j);
                abuf[nxt][j] = *(const v2f*)(arow + k);
                bbuf[nxt][j] = *(const v2f*)(brow + k);
            }
        }
        #pragma unroll
        for (int j = 0; j < CH; ++j)
            acc = wmma_f32(abuf[cur][j], bbuf[cur][j], acc);
    }

    int n = n0 + lm;
    int mq_n = tmask[bIdx * LL + n];
    #pragma unroll
    for (int r = 0; r < 8; ++r) {
        int m = m0 + r + 8 * hi;
        // time term: dot(TM[b,m,n,:], q[b,m,:]) -- streams the 134MB tensor.
        // Regular-temporal loads: TM is re-read in layer 2 and fits in L2 (192MB).
        const float4* tmr = (const float4*)(tm + (((size_t)bIdx * LL + m) * LL + n) * DD);
        const float4* qr  = (const float4*)(qb + (size_t)m * DD);
        float dot = 0.0f;
        #pragma unroll
        for (int d = 0; d < DD / 4; ++d) {
            float4 t = tmr[d];
            float4 qv = qr[d];
            dot += t.x * qv.x + t.y * qv.y + t.z * qv.z + t.w * qv.w;
        }
        float val = (acc[r] + dot) * SCALEV;
        bool msk = (tmask[bIdx * LL + m] != 0) || (mq_n != 0) || (n > m);
        scores[((size_t)bIdx * LL + m) * LL + n] = msk ? NEGV : val;
    }
}

// -----------------------------------------------------------------------------
extern "C" void kernel_launch(void* const* d_in, const int* in_sizes, int n_in,
                              void* d_out, int out_size, void* d_ws, size_t ws_size,
                              hipStream_t stream) {
    (void)in_sizes; (void)n_in; (void)out_size; (void)ws_size;

    const float* embeds = (const float*)d_in[0];
    const float* labels = (const float*)d_in[1];
    const int*   tmask  = (const int*)d_in[2];
    const float* TM     = (const float*)d_in[3];
    const float* Wq = (const float*)d_in[4];  const float* bq = (const float*)d_in[5];
    const float* Wk = (const float*)d_in[6];  const float* bk = (const float*)d_in[7];
    const float* Wv = (const float*)d_in[8];  const float* bv = (const float*)d_in[9];
    const float* ln1_g = (const float*)d_in[10]; const float* ln1_b = (const float*)d_in[11];
    const float* ln2_g = (const float*)d_in[12]; const float* ln2_b = (const float*)d_in[13];
    const float* c1w = (const float*)d_in[14]; const float* c1b = (const float*)d_in[15];
    const float* c2w = (const float*)d_in[16]; const float* c2b = (const float*)d_in[17];
    const float* last_g = (const float*)d_in[18]; const float* last_b = (const float*)d_in[19];
    const float* Wpoi = (const float*)d_in[20]; const float* bpoi = (const float*)d_in[21];
    const float* Wcat = (const float*)d_in[22]; const float* bcat = (const float*)d_in[23];

    float* ws = (float*)d_ws;
    const size_t T = (size_t)ROWS * DD;   // 262144 floats = 1 MB
    float* seqs  = ws + 0 * T;
    float* Qln   = ws + 1 * T;
    float* qbuf  = ws + 2 * T;   // reused as FFN hidden h
    float* kbuf  = ws + 3 * T;
    float* vbuf  = ws + 4 * T;
    float* sc    = ws + 5 * T;
    float* tmp   = ws + 6 * T;
    float* seqs2 = ws + 7 * T;
    float* logf  = ws + 8 * T;

    hipMemcpyAsync(seqs, embeds, T * sizeof(float), hipMemcpyDeviceToDevice, stream);

    const dim3 blk256(256);
    const dim3 gM(ROWS / 16, 1);          // N=128 GEMMs: 128 row-tiles, 1 col-block

    for (int i = 0; i < NLAYERS; ++i) {
        size_t od = (size_t)i * DD * DD;
        size_t ov = (size_t)i * DD;

        // Q = LN1(seqs)
        ln_kernel<false><<<ROWS, 32, 0, stream>>>(seqs, ln1_g + ov, ln1_b + ov, nullptr, Qln);
        // q,k,v projections:  N=128, +bias
        gemm_kernel<DD, true, false, false, false, false, false><<<gM, blk256, 0, stream>>>(
            Qln, Wq + od, bq + ov, nullptr, nullptr, qbuf);
        gemm_kernel<DD, true, false, false, false, false, false><<<gM, blk256, 0, stream>>>(
            Qln, Wk + od, bk + ov, nullptr, nullptr, kbuf);
        gemm_kernel<DD, true, false, false, false, false, false><<<gM, blk256, 0, stream>>>(
            Qln, Wv + od, bv + ov, nullptr, nullptr, vbuf);
        // masked scaled scores (qk^T + time-matrix term)
        scores_kernel<<<dim3(BATCH, LL / 16), blk256, 0, stream>>>(qbuf, kbuf, TM, tmask, sc);
        softmax_kernel<<<ROWS, 32, 0, stream>>>(sc);
        // seqs_pre = Q + attn @ v   (per-batch B, +residual)
        gemm_kernel<DD, false, false, true, false, true, false><<<gM, blk256, 0, stream>>>(
            sc, vbuf, nullptr, Qln, nullptr, tmp);
        // seqs2 = LN2(seqs_pre)
        ln_kernel<false><<<ROWS, 32, 0, stream>>>(tmp, ln2_g + ov, ln2_b + ov, nullptr, seqs2);
        // h = relu(seqs2 @ c1 + b)
        gemm_kernel<DD, true, true, false, false, false, false><<<gM, blk256, 0, stream>>>(
            seqs2, c1w + od, c1b + ov, nullptr, nullptr, qbuf);
        // seqs = (seqs2 + h @ c2 + b) * keep
        gemm_kernel<DD, true, false, true, true, false, false><<<gM, blk256, 0, stream>>>(
            qbuf, c2w + od, c2b + ov, seqs2, tmask, seqs);
    }

    // log_feats = LN(seqs)*g+b + labels_timeInterval
    ln_kernel<true><<<ROWS, 32, 0, stream>>>(seqs, last_g, last_b, labels, logf);

    float* out = (float*)d_out;
    // out_poi: 41MB written once, never re-read -> non-temporal stores.
    gemm_kernel<NUM_POI, true, false, false, false, false, true>
        <<<dim3(ROWS / 16, (NUM_POI + 127) / 128), blk256, 0, stream>>>(
            logf, Wpoi, bpoi, nullptr, nullptr, out);
    gemm_kernel<NUM_CAT, true, false, false, false, false, false>
        <<<dim3(ROWS / 16, (NUM_CAT + 127) / 128), blk256, 0, stream>>>(
            logf, Wcat, bcat, nullptr, nullptr, out + (size_t)ROWS * NUM_POI);
}